// LinearAttention_80298708566521
// MI455X (gfx1250) — compile-verified
//
#include <hip/hip_runtime.h>

// ---------------------------------------------------------------------------
// Linear attention for MI455X (gfx1250), fp32 WMMA path (V_WMMA_F32_16X16X4_F32)
// b=8, c=64, h*w = n = 65536, HEADS=4, HEAD_DIM=16
// ---------------------------------------------------------------------------

typedef __attribute__((ext_vector_type(2))) float v2f;
typedef __attribute__((ext_vector_type(8))) float v8f;

#define N_TOK   65536
#define CCH     64
#define HEADS   4
#define WAVES   8          // 256-thread workgroup = 8 wave32
#define LSTRIDE 18         // padded LDS row stride (floats): conflict-free + 8B aligned

__device__ __forceinline__ float phi_f(float x) {
    // elu(x)+1  ==  x+1 (x>=0),  exp(x) (x<0)
    return x > 0.0f ? (x + 1.0f) : __expf(x);
}

__device__ __forceinline__ v8f wmma4(v2f a, v2f b, v8f c) {
    // D(16x16) = A(16x4) x B(4x16) + C, fp32
    return __builtin_amdgcn_wmma_f32_16x16x4_f32(
        /*neg_a=*/false, a, /*neg_b=*/false, b,
        /*c_mod=*/(short)0, c, /*reuse_a=*/false, /*reuse_b=*/false);
}

// A-fragment of a 64x64 row-major weight matrix, row-block r (rows 16r..16r+15),
// K-step kk (cols 4kk..4kk+3).  lane l: A[l%16][4kk + 2*(l/16) + j]
__device__ __forceinline__ v2f load_wfrag(const float* __restrict__ W,
                                          int r, int kk, int lm, int half) {
    const float* p = W + (r * 16 + lm) * CCH + kk * 4 + half * 2;
    v2f a;
    a.x = p[0];
    a.y = p[1];
    return a;
}

__global__ __launch_bounds__(256) void la_zero_kernel(float* __restrict__ p, int n) {
    int i = blockIdx.x * 256 + threadIdx.x;
    if (i < n) p[i] = 0.0f;
}

// ---------------------------------------------------------------------------
// Kernel 1: K/V projection + phi(K), accumulate KV[b][h][d][e] = sum_n k*v
// ---------------------------------------------------------------------------
__global__ __launch_bounds__(256) void la_kv_kernel(
    const float* __restrict__ x, const float* __restrict__ wk,
    const float* __restrict__ wv, float* __restrict__ kv) {

    __shared__ float k_lds[WAVES][16 * LSTRIDE];
    __shared__ float v_lds[WAVES][16 * LSTRIDE];

    const int wid  = threadIdx.x >> 5;
    const int lane = threadIdx.x & 31;
    const int half = lane >> 4;
    const int lm   = lane & 15;
    const int b    = blockIdx.y;
    const float* X = x + (size_t)b * CCH * N_TOK;

    v8f zero = {};
    v8f kvacc[HEADS];
#pragma unroll
    for (int h = 0; h < HEADS; ++h) kvacc[h] = zero;

    const int wave_global     = blockIdx.x * WAVES + wid;
    const int waves_per_batch = gridDim.x * WAVES;

    for (int tile = wave_global; tile < N_TOK / 16; tile += waves_per_batch) {
        const int t = tile * 16 + lm;

        // X tile as B-fragments: xb[kk] lane l = X[4kk+2*half+j][token lm]
        v2f xb[16];
#pragma unroll
        for (int kk = 0; kk < 16; ++kk) {
            const int c0 = kk * 4 + half * 2;
            xb[kk].x = X[(size_t)c0 * N_TOK + t];
            xb[kk].y = X[(size_t)(c0 + 1) * N_TOK + t];
        }

#pragma unroll
        for (int h = 0; h < HEADS; ++h) {
            v8f kacc = zero, vacc = zero;
#pragma unroll
            for (int kk = 0; kk < 16; ++kk) {
                kacc = wmma4(load_wfrag(wk, h, kk, lm, half), xb[kk], kacc);
                vacc = wmma4(load_wfrag(wv, h, kk, lm, half), xb[kk], vacc);
            }
#pragma unroll
            for (int i = 0; i < 8; ++i) kacc[i] = phi_f(kacc[i]);

            __syncthreads();   // protect LDS reuse (uniform trip counts)
#pragma unroll
            for (int i = 0; i < 8; ++i) {
                k_lds[wid][(i + half * 8) * LSTRIDE + lm] = kacc[i];
                v_lds[wid][(i + half * 8) * LSTRIDE + lm] = vacc[i];
            }
            __syncthreads();

            // KV += K(16d x 16tok) * V^T(16tok x 16e), K-dim = tokens, 4 steps
#pragma unroll
            for (int kk = 0; kk < 4; ++kk) {
                v2f ka = *(const v2f*)&k_lds[wid][lm * LSTRIDE + kk * 4 + half * 2];
                v2f vb = *(const v2f*)&v_lds[wid][lm * LSTRIDE + kk * 4 + half * 2];
                kvacc[h] = wmma4(ka, vb, kvacc[h]);
            }
        }
    }

    // flush per-wave accumulators: KV[b][h][d][e], d = i+8*half, e = lm
    float* kvb = kv + (size_t)b * HEADS * 256;
#pragma unroll
    for (int h = 0; h < HEADS; ++h)
#pragma unroll
        for (int i = 0; i < 8; ++i)
            atomicAdd(&kvb[h * 256 + (i + half * 8) * 16 + lm], kvacc[h][i]);
}

// ---------------------------------------------------------------------------
// Kernel 2: q = phi(Wq x);  out_h = KV_h^T q_h;  y = Wo out + bo
// ---------------------------------------------------------------------------
__global__ __launch_bounds__(256) void la_out_kernel(
    const float* __restrict__ x, const float* __restrict__ wq,
    const float* __restrict__ wo, const float* __restrict__ bo,
    const float* __restrict__ kv, float* __restrict__ out) {

    __shared__ float t_lds[WAVES][64 * LSTRIDE];   // q tile, later out_pre tile
    __shared__ float kv_s[HEADS * 256];

    const int wid  = threadIdx.x >> 5;
    const int lane = threadIdx.x & 31;
    const int half = lane >> 4;
    const int lm   = lane & 15;
    const int b    = blockIdx.y;
    const float* X = x + (size_t)b * CCH * N_TOK;
    float*       Y = out + (size_t)b * CCH * N_TOK;

    for (int i = threadIdx.x; i < HEADS * 256; i += 256)
        kv_s[i] = kv[(size_t)b * HEADS * 256 + i];

    v8f zero = {};
    const int wave_global     = blockIdx.x * WAVES + wid;
    const int waves_per_batch = gridDim.x * WAVES;

    for (int tile = wave_global; tile < N_TOK / 16; tile += waves_per_batch) {
        const int t = tile * 16 + lm;

        v2f xb[16];
#pragma unroll
        for (int kk = 0; kk < 16; ++kk) {
            const int c0 = kk * 4 + half * 2;
            xb[kk].x = X[(size_t)c0 * N_TOK + t];
            xb[kk].y = X[(size_t)(c0 + 1) * N_TOK + t];
        }

        __syncthreads();   // previous iteration done reading t_lds (+ kv_s load)

        // q projection, phi, stage rows [0..64) of t_lds
#pragma unroll
        for (int r = 0; r < 4; ++r) {
            v8f qacc = zero;
#pragma unroll
            for (int kk = 0; kk < 16; ++kk)
                qacc = wmma4(load_wfrag(wq, r, kk, lm, half), xb[kk], qacc);
#pragma unroll
            for (int i = 0; i < 8; ++i)
                t_lds[wid][(r * 16 + i + half * 8) * LSTRIDE + lm] = phi_f(qacc[i]);
        }
        __syncthreads();

        // out_h[e,tok] = sum_d KV[d][e] * q_h[d][tok]   (K-dim = d, 4 steps)
        v8f outd[HEADS];
#pragma unroll
        for (int h = 0; h < HEADS; ++h) {
            v8f acc = zero;
#pragma unroll
            for (int kk = 0; kk < 4; ++kk) {
                const int d0 = kk * 4 + half * 2;
                v2f a;   // A[m=e][k=d] = KV[d][e]
                a.x = kv_s[h * 256 + (d0 + 0) * 16 + lm];
                a.y = kv_s[h * 256 + (d0 + 1) * 16 + lm];
                v2f bq;  // B[k=d][n=tok] = q_h[d][tok]
                bq.x = t_lds[wid][(h * 16 + d0 + 0) * LSTRIDE + lm];
                bq.y = t_lds[wid][(h * 16 + d0 + 1) * LSTRIDE + lm];
                acc = wmma4(a, bq, acc);
            }
            outd[h] = acc;
        }
        __syncthreads();
#pragma unroll
        for (int h = 0; h < HEADS; ++h)
#pragma unroll
            for (int i = 0; i < 8; ++i)
                t_lds[wid][(h * 16 + i + half * 8) * LSTRIDE + lm] = outd[h][i];
        __syncthreads();

        // y = Wo * out_pre + bo (bias folded into C accumulator init)
#pragma unroll
        for (int r = 0; r < 4; ++r) {
            v8f acc;
#pragma unroll
            for (int i = 0; i < 8; ++i) acc[i] = bo[r * 16 + i + half * 8];
#pragma unroll
            for (int kk = 0; kk < 16; ++kk) {
                v2f a = load_wfrag(wo, r, kk, lm, half);
                const int c0 = kk * 4 + half * 2;
                v2f bfr;
                bfr.x = t_lds[wid][(c0 + 0) * LSTRIDE + lm];
                bfr.y = t_lds[wid][(c0 + 1) * LSTRIDE + lm];
                acc = wmma4(a, bfr, acc);
            }
#pragma unroll
            for (int i = 0; i < 8; ++i)
                Y[(size_t)(r * 16 + i + half * 8) * N_TOK + t] = acc[i];
        }
    }
}

// ---------------------------------------------------------------------------
extern "C" void kernel_launch(void* const* d_in, const int* in_sizes, int n_in,
                              void* d_out, int out_size, void* d_ws, size_t ws_size,
                              hipStream_t stream) {
    const float* x  = (const float*)d_in[0];
    const float* wq = (const float*)d_in[1];
    const float* wk = (const float*)d_in[2];
    const float* wv = (const float*)d_in[3];
    const float* wo = (const float*)d_in[4];
    const float* bo = (const float*)d_in[5];
    float* out = (float*)d_out;
    float* kv  = (float*)d_ws;   // 8 * 4 * 16 * 16 floats = 32 KB

    const int kv_elems = 8 * HEADS * 256;
    la_zero_kernel<<<dim3((kv_elems + 255) / 256), 256, 0, stream>>>(kv, kv_elems);

    // 4096 16-token tiles per batch; 256 WGs x 8 waves -> 2 tiles per wave (uniform)
    la_kv_kernel<<<dim3(256, 8), 256, 0, stream>>>(x, wk, wv, kv);
    la_out_kernel<<<dim3(256, 8), 256, 0, stream>>>(x, wq, wo, bo, kv, out);
}